// QNetwork_28587302323119
// MI455X (gfx1250) — compile-verified
//
#include <hip/hip_runtime.h>
#include <cstddef>
#include <cstdint>

// ---------------------------------------------------------------------------
// QNetwork (3x BiLSTM + additive attention + dueling head) for MI455X gfx1250.
//
// Roofline: ~150 MB fp32 weights fit in the 192 MB L2 (one-time HBM cost
// ~6.5us at 23.3 TB/s); total ~82 GFLOP. The 3*256-step LSTM recurrence
// serializes the work, so the design minimizes per-step latency:
//  - Whh slices LDS-resident across all 256 steps (loaded once per layer),
//  - ds_load_b128 dot products (stride 772 floats: 16B-aligned, <=3-way bank
//    conflict),
//  - one-atomic grid barrier + ping-pong h buffer (single barrier per step).
// All time-parallel GEMMs (~70% of FLOPs) use V_WMMA_F32_16X16X4_F32
// (fp32-exact vs the reference), software-pipelined with double-buffered
// register chunks so ~20 loads are in flight per wait instead of 1.
// The attention kernel stages hq rows into LDS via the Tensor Data Mover
// (tensor_load_to_lds + s_wait_tensorcnt). Workspace use: ~46 MB.
// ---------------------------------------------------------------------------

constexpr int NB   = 4;       // batch
constexpr int NS1  = 256;     // states seq len
constexpr int NS2  = 64;      // actions seq len
constexpr int NH   = 768;     // hidden
constexpr int NH2  = 2 * NH;  // 1536
constexpr int NG   = 4 * NH;  // 3072 gate rows
constexpr int NLAB = 3;

typedef __attribute__((ext_vector_type(2))) float v2f;
typedef __attribute__((ext_vector_type(8))) float v8f;
typedef __attribute__((ext_vector_type(4))) unsigned int u32x4;
typedef __attribute__((ext_vector_type(4))) int i32x4;
typedef __attribute__((ext_vector_type(8))) int i32x8;

#if defined(__HIP_DEVICE_COMPILE__) && __has_builtin(__builtin_amdgcn_tensor_load_to_lds)
#define HAVE_TDM 1
#else
#define HAVE_TDM 0
#endif

// ========================= fp32 WMMA GEMM ==================================
// Y[m][n] = sum_k X[m][k] * W[n][koff+k]  (+ bias[n])
// A tile 16x4 f32, B tile 4x16 f32 (B[k][n] = W[n][k]), C/D 16x16 f32.
// One wave owns a 16(M) x 64(N) tile (4 accumulators); K is consumed in
// 16-deep chunks, double-buffered in registers so the 20 loads of the next
// chunk are in flight while the current chunk's 16 WMMAs execute.
// Requires K % 32 == 0 (all call sites use K in {768, 1536}).
__global__ __launch_bounds__(256)
void gemm_wmma_f32(const float* __restrict__ X, int ldx,
                   const float* __restrict__ W, int ldw, int koff,
                   const float* __restrict__ bias,
                   float* __restrict__ Y, int ldy,
                   int M, int N, int K)
{
  const int ntn   = N >> 6;                        // 64-wide column tiles
  const int total = (M >> 4) * ntn;
  const int w     = blockIdx.x * 8 + (threadIdx.x >> 5);
  if (w >= total) return;                          // wave-uniform: EXEC stays all-1s
  const int tm = w / ntn, tn = w - tm * ntn;
  const int m0 = tm << 4, n0 = tn << 6;
  const int lane = threadIdx.x & 31;
  const int mr   = lane & 15;                      // M (for A) / N (for B) index
  const int kb   = (lane >> 4) << 1;               // K pair base: 0 or 2
  const int hi   = (lane >> 4) << 3;               // C/D row group: 0 or 8

  const v8f vz = {0.f,0.f,0.f,0.f,0.f,0.f,0.f,0.f};
  v8f acc[4] = {vz, vz, vz, vz};

  const float* xrow  = X + (size_t)(m0 + mr) * ldx;
  const float* wbase = W + (size_t)koff;

  auto loadA = [&](int k0, float2 (&a)[4]) {
#pragma unroll
    for (int i = 0; i < 4; ++i)
      a[i] = *(const float2*)(xrow + k0 + (i << 2) + kb);
  };
  auto loadB = [&](int k0, float2 (&b)[4][4]) {
#pragma unroll
    for (int s = 0; s < 4; ++s) {
      const float* wp = wbase + (size_t)(n0 + (s << 4) + mr) * ldw + k0 + kb;
#pragma unroll
      for (int i = 0; i < 4; ++i)
        b[s][i] = *(const float2*)(wp + (i << 2));
    }
  };
  auto domma = [&](float2 (&a)[4], float2 (&b)[4][4]) {
#pragma unroll
    for (int i = 0; i < 4; ++i) {
      v2f av; av.x = a[i].x; av.y = a[i].y;
#pragma unroll
      for (int s = 0; s < 4; ++s) {
        v2f bv; bv.x = b[s][i].x; bv.y = b[s][i].y;
        acc[s] = __builtin_amdgcn_wmma_f32_16x16x4_f32(false, av, false, bv,
                                                       (short)0, acc[s],
                                                       false, false);
      }
    }
  };

  float2 aA[4], aB[4], bA[4][4], bB[4][4];
  loadA(0, aA); loadB(0, bA);
  for (int k0 = 0; k0 < K; k0 += 32) {
    loadA(k0 + 16, aB); loadB(k0 + 16, bB);        // k0+16 < K since K%32==0
    domma(aA, bA);
    if (k0 + 32 < K) { loadA(k0 + 32, aA); loadB(k0 + 32, bA); }
    domma(aB, bB);
  }

#pragma unroll
  for (int s = 0; s < 4; ++s) {
    const int n  = n0 + (s << 4) + mr;
    const float bn = bias ? bias[n] : 0.f;
#pragma unroll
    for (int r = 0; r < 8; ++r)
      Y[(size_t)(m0 + hi + r) * ldy + n] = acc[s][r] + bn;
  }
}

// ========================= LSTM recurrence =================================
// Persistent kernel; 64 WGs per direction, each owns 12 hidden units
// (48 gate rows). Whh slice lives in LDS for all 256 steps. Row stride 772
// floats: 3088 B (16B aligned -> ds_load_b128) and 772 % 64 == 4, so at a
// given k only rows {r, r+16, r+32} collide (3-way worst case). Grid barrier
// is one atomic per WG per step; h ping-pongs in global memory so a single
// barrier per step is race-free.
constexpr int NWGD = 64;              // workgroups per direction
constexpr int HS   = NH / NWGD;       // 12 hidden units per WG
constexpr int NRW  = 4 * HS;          // 48 gate rows per WG
constexpr int HP   = NH + 4;          // padded LDS row stride (floats)

__device__ __forceinline__ float sigmoidf_(float x) {
  return 1.f / (1.f + __expf(-x));
}

__global__ __launch_bounds__(256, 1)
void lstm_recur(const float* __restrict__ Xg,    // [2][NB*NS1][NG] x@Wih.T+b
                const float* __restrict__ Whh,   // [2][NG][NH]
                float* __restrict__ Y,           // [NB][NS1][NH2]
                float* __restrict__ hglob,       // [2(pingpong)][2(dir)][NB][NH]
                unsigned* __restrict__ sync, int apply_tanh)
{
  __shared__ float wlds[NRW * HP];     // 148.2 KB
  __shared__ float hprev[NB * HP];     // 12.4 KB
  __shared__ float gbuf[NB * NRW];
  __shared__ float cst[NB * HS];

  const int wg    = blockIdx.x;
  const int dir   = wg / NWGD;
  const int slice = wg - dir * NWGD;
  const int j0    = slice * HS;
  const int tid   = threadIdx.x;
  const float* Wd = Whh + (size_t)dir * NG * NH;
  const float* Xd = Xg  + (size_t)dir * (NB * NS1) * NG;

  // Load the Whh slice once; it stays resident for all 256 steps.
  for (int r = 0; r < NRW; ++r) {
    const int g = r / HS, u = r - g * HS;
    const float* src = Wd + (size_t)(g * NH + j0 + u) * NH;
    for (int k = tid; k < NH; k += 256) wlds[r * HP + k] = src[k];
  }
  for (int i = tid; i < NB * HS; i += 256) cst[i] = 0.f;
  for (int i = tid; i < NB * HP; i += 256) hprev[i] = 0.f;
  __syncthreads();

  const unsigned nwg = gridDim.x;
  for (int s = 0; s < NS1; ++s) {
    const int t = dir ? (NS1 - 1 - s) : s;

    if (tid < NB * NRW) {                         // 192 dots of length 768
      const int b = tid / NRW, r = tid - b * NRW;
      const float4* wrow = (const float4*)(wlds + r * HP);
      const float4* hrow = (const float4*)(hprev + b * HP);
      float acc = 0.f;
#pragma unroll 4
      for (int k = 0; k < NH / 4; ++k) {
        const float4 wv = wrow[k];
        const float4 hv = hrow[k];
        acc = fmaf(wv.x, hv.x, acc);
        acc = fmaf(wv.y, hv.y, acc);
        acc = fmaf(wv.z, hv.z, acc);
        acc = fmaf(wv.w, hv.w, acc);
      }
      const int g = r / HS, u = r - g * HS;
      acc += Xd[(size_t)(b * NS1 + t) * NG + g * NH + j0 + u];
      gbuf[tid] = acc;                            // tid == b*NRW + r
    }
    __syncthreads();

    float* hw = hglob + (size_t)(((s & 1) * 2 + dir) * NB * NH);
    if (tid < NB * HS) {
      const int b = tid / HS, u = tid - b * HS;
      const float gi = gbuf[b * NRW + 0 * HS + u];
      const float gf = gbuf[b * NRW + 1 * HS + u];
      const float gg = gbuf[b * NRW + 2 * HS + u];
      const float go = gbuf[b * NRW + 3 * HS + u];
      const float c  = sigmoidf_(gf) * cst[tid] + sigmoidf_(gi) * tanhf(gg);
      cst[tid] = c;
      const float h = sigmoidf_(go) * tanhf(c);
      hw[b * NH + j0 + u] = h;                    // raw h feeds the recurrence
      Y[(size_t)(b * NS1 + t) * NH2 + dir * NH + j0 + u] =
          apply_tanh ? tanhf(h) : h;
    }
    __threadfence();
    __syncthreads();

    if (tid == 0) {                               // grid barrier (arrive+wait)
      atomicAdd(sync, 1u);
      const unsigned target = (unsigned)(s + 1) * nwg;
      while (__atomic_load_n(sync, __ATOMIC_ACQUIRE) < target)
        __builtin_amdgcn_s_sleep(2);
    }
    __syncthreads();

    if (s + 1 < NS1) {                            // pull the full h for this step
      const volatile float* hr =
          (const volatile float*)(hglob + (size_t)(((s & 1) * 2 + dir) * NB * NH));
      for (int i = tid; i < NB * NH; i += 256) {
        const int b = i / NH, k = i - b * NH;
        hprev[b * HP + k] = hr[i];
      }
      __syncthreads();
    }
  }
}

// ========================= attention: scores + softmax =====================
// One block per (b,q); thread k handles state k. P = A / clip(sum A, 2e-15).
// The hq row is DMA'd into LDS by the Tensor Data Mover (2-D D#: data_size=4B,
// tile 768x1, type=2 "image"); the wave then waits on TENSORcnt.
__global__ __launch_bounds__(256)
void attn_scores(const float* __restrict__ hq,   // [NB*NS2][NH]
                 const float* __restrict__ hk,   // [NB*NS1][NH]
                 const float* __restrict__ b1,   // [NH]
                 const float* __restrict__ w2,   // [NH]
                 const float* __restrict__ b2,   // [1]
                 const float* __restrict__ amask,// [NB][NS2]
                 const float* __restrict__ smask,// [NB][NS1]
                 float* __restrict__ P)          // [NB*NS2][NS1]
{
  __shared__ float qv[NH];
  __shared__ float w2s[NH];
  __shared__ float red[256];
  const int bq = blockIdx.x;
  const int b  = bq / NS2, q = bq - b * NS2;
  const int tid = threadIdx.x;

#if HAVE_TDM
  if ((tid >> 5) == 0) {                          // wave 0 issues the DMA
    const unsigned long long ga =
        (unsigned long long)(uintptr_t)(hq + (size_t)bq * NH);
    const unsigned ldsoff = (unsigned)(uintptr_t)(&qv[0]); // low 32b = LDS offset
    u32x4 g0;
    g0[0] = 1u;                                   // count=1, user descriptor
    g0[1] = ldsoff;                               // lds_addr (bytes)
    g0[2] = (unsigned)(ga & 0xFFFFFFFFu);         // global_addr[31:0]
    g0[3] = (unsigned)((ga >> 32) & 0x01FFFFFFu)  // global_addr[56:32]
            | 0x80000000u;                        // type=2 ("image")
    i32x8 g1;
    g1[0] = (int)(2u << 16);                      // data_size=4B, mask/pad/iter=0
    g1[1] = (int)(768u << 16);                    // tensor_dim0[15:0]=768
    g1[2] = (int)(1u << 16);                      // tensor_dim0 hi=0, tensor_dim1=1
    g1[3] = (int)(768u << 16);                    // tile_dim0=768
    g1[4] = 1;                                    // tile_dim1=1, tile_dim2=0
    g1[5] = 768;                                  // tensor_dim0_stride=768
    g1[6] = 0; g1[7] = 0;
    const i32x4 gz4 = {0, 0, 0, 0};               // dims 2-4 unused
    const i32x8 gz8 = {0, 0, 0, 0, 0, 0, 0, 0};   // trailing group unused
    __builtin_amdgcn_tensor_load_to_lds(g0, g1, gz4, gz4, gz8, 0);
    __builtin_amdgcn_s_wait_tensorcnt(0);
  }
  for (int e = tid; e < NH; e += 256) w2s[e] = w2[e];
  __syncthreads();
  for (int e = tid; e < NH; e += 256) qv[e] += b1[e];
  __syncthreads();
#else
  for (int e = tid; e < NH; e += 256) {
    qv[e]  = hq[(size_t)bq * NH + e] + b1[e];
    w2s[e] = w2[e];
  }
  __syncthreads();
#endif

  const int k = tid;                              // NS1 == blockDim
  const float* kr = hk + (size_t)(b * NS1 + k) * NH;
  float acc = 0.f;
  for (int e = 0; e < NH; ++e) {
    float v = qv[e] + kr[e];
    v = v > 0.f ? v : 0.f;
    acc = fmaf(v, w2s[e], acc);
  }
  const float score = acc + b2[0];
  const float m = amask[b * NS2 + q] * smask[b * NS1 + k];
  const float A = (m == 0.f) ? 0.f : __expf(score);

  red[tid] = A;
  __syncthreads();
  for (int off = 128; off > 0; off >>= 1) {
    if (tid < off) red[tid] += red[tid + off];
    __syncthreads();
  }
  float ssum = red[0];
  ssum = ssum < 2e-15f ? 2e-15f : ssum;
  P[(size_t)bq * NS1 + k] = A / ssum;
}

// ========================= states_feat = P @ out ===========================
__global__ __launch_bounds__(256)
void states_feat_k(const float* __restrict__ P,    // [NB*NS2][NS1]
                   const float* __restrict__ Yout, // [NB*NS1][NH2]
                   float* __restrict__ SF)         // [NB*NS2][NH2]
{
  __shared__ float Ps[NS1];
  const int bq = blockIdx.x;
  const int b  = bq / NS2;
  const int tid = threadIdx.x;
  Ps[tid] = P[(size_t)bq * NS1 + tid];
  __syncthreads();
  for (int d = tid; d < NH2; d += 256) {
    float acc = 0.f;
    for (int k = 0; k < NS1; ++k)
      acc = fmaf(Ps[k], Yout[(size_t)(b * NS1 + k) * NH2 + d], acc);
    SF[(size_t)bq * NH2 + d] = acc;
  }
}

// ===== adv[b,q,l] = sum_h tmp[bq][l*NH+h] * actions[bq][h] + bias[l] =======
__global__ __launch_bounds__(256)
void adv_reduce(const float* __restrict__ tmp,     // [NB*NS2][NLAB*NH]
                const float* __restrict__ actions, // [NB*NS2][NH]
                const float* __restrict__ bias,    // [NLAB]
                float* __restrict__ adv)           // [NB*NS2][NLAB]
{
  __shared__ float r0[256], r1[256], r2[256];
  const int bq = blockIdx.x;
  const int tid = threadIdx.x;
  const float* ar = actions + (size_t)bq * NH;
  const float* tr = tmp + (size_t)bq * (NLAB * NH);
  float a0 = 0.f, a1 = 0.f, a2 = 0.f;
  for (int h = tid; h < NH; h += 256) {
    const float av = ar[h];
    a0 = fmaf(tr[0 * NH + h], av, a0);
    a1 = fmaf(tr[1 * NH + h], av, a1);
    a2 = fmaf(tr[2 * NH + h], av, a2);
  }
  r0[tid] = a0; r1[tid] = a1; r2[tid] = a2;
  __syncthreads();
  for (int off = 128; off > 0; off >>= 1) {
    if (tid < off) { r0[tid] += r0[tid+off]; r1[tid] += r1[tid+off]; r2[tid] += r2[tid+off]; }
    __syncthreads();
  }
  if (tid == 0) {
    adv[bq * NLAB + 0] = r0[0] + bias[0];
    adv[bq * NLAB + 1] = r1[0] + bias[1];
    adv[bq * NLAB + 2] = r2[0] + bias[2];
  }
}

// ===== q = val + adv - mean(adv);  val = (sum_{q,d} SF*valW)/num + valb ====
__global__ __launch_bounds__(256)
void finalize_k(const float* __restrict__ SF,     // [NB*NS2][NH2]
                const float* __restrict__ adv,    // [NB*NS2][NLAB]
                const float* __restrict__ amask,  // [NB][NS2]
                const float* __restrict__ valW,   // [NH2]
                const float* __restrict__ valb,   // [1]
                float* __restrict__ out)          // [NB*NS2][NLAB]
{
  __shared__ float r0[256], r1[256], r2[256];
  const int b = blockIdx.x;
  const int tid = threadIdx.x;

  float vacc = 0.f;
  for (int i = tid; i < NS2 * NH2; i += 256) {
    const int d = i % NH2;
    vacc = fmaf(SF[(size_t)(b * NS2) * NH2 + i], valW[d], vacc);
  }
  float nsum = 0.f;
  for (int q = tid; q < NS2; q += 256) nsum += amask[b * NS2 + q];
  float asum = 0.f;
  for (int i = tid; i < NS2 * NLAB; i += 256) asum += adv[(size_t)(b * NS2) * NLAB + i];

  r0[tid] = vacc; r1[tid] = nsum; r2[tid] = asum;
  __syncthreads();
  for (int off = 128; off > 0; off >>= 1) {
    if (tid < off) { r0[tid] += r0[tid+off]; r1[tid] += r1[tid+off]; r2[tid] += r2[tid+off]; }
    __syncthreads();
  }
  const float val  = r0[0] / r1[0] + valb[0];
  const float madv = r2[0] / (float)(NS2 * NLAB);
  for (int i = tid; i < NS2 * NLAB; i += 256) {
    const size_t idx = (size_t)(b * NS2) * NLAB + i;
    out[idx] = val + adv[idx] - madv;
  }
}

// ========================= host side =======================================
static inline void run_gemm(hipStream_t st, const float* X, int ldx,
                            const float* W, int ldw, int koff, const float* bias,
                            float* Y, int ldy, int M, int N, int K)
{
  const int tiles  = (M / 16) * (N / 64);
  const int blocks = (tiles + 7) / 8;
  gemm_wmma_f32<<<blocks, 256, 0, st>>>(X, ldx, W, ldw, koff, bias, Y, ldy, M, N, K);
}

extern "C" void kernel_launch(void* const* d_in, const int* in_sizes, int n_in,
                              void* d_out, int out_size, void* d_ws, size_t ws_size,
                              hipStream_t stream)
{
  (void)in_sizes; (void)n_in; (void)out_size; (void)ws_size;
  const float* states  = (const float*)d_in[0];
  const float* smask   = (const float*)d_in[1];
  const float* actions = (const float*)d_in[2];
  const float* amask   = (const float*)d_in[3];
  const float* Wih0    = (const float*)d_in[4];
  const float* Whh0    = (const float*)d_in[5];
  const float* b0      = (const float*)d_in[6];
  const float* Wih12   = (const float*)d_in[7];
  const float* Whh12   = (const float*)d_in[8];
  const float* b12     = (const float*)d_in[9];
  const float* attnW1  = (const float*)d_in[10];
  const float* attnb1  = (const float*)d_in[11];
  const float* attnW2  = (const float*)d_in[12];
  const float* attnb2  = (const float*)d_in[13];
  const float* valW    = (const float*)d_in[14];
  const float* valb    = (const float*)d_in[15];
  const float* weight  = (const float*)d_in[16];
  const float* bias    = (const float*)d_in[17];
  float* out = (float*)d_out;

  // Workspace carve-up (~46 MB).
  char* ws = (char*)d_ws;
  unsigned* sync = (unsigned*)ws;                       // 3 counters (256 B)
  float* Xg  = (float*)(ws + 256);                      // [2][1024][3072]
  float* Y0  = Xg  + (size_t)2 * 1024 * NG;             // [1024][1536]
  float* Y1  = Y0  + (size_t)1024 * NH2;                // [1024][1536]
  float* hgl = Y1  + (size_t)1024 * NH2;                // [2][2][NB][NH]
  float* hq  = hgl + (size_t)2 * 2 * NB * NH;           // [256][768]
  float* hk  = hq  + (size_t)NB * NS2 * NH;             // [1024][768]
  float* P   = hk  + (size_t)NB * NS1 * NH;             // [256][256]
  float* SF  = P   + (size_t)NB * NS2 * NS1;            // [256][1536]
  float* tmp = SF  + (size_t)NB * NS2 * NH2;            // [256][2304]
  float* adv = tmp + (size_t)NB * NS2 * NLAB * NH;      // [256][3]

  (void)hipMemsetAsync(sync, 0, 256, stream);           // reset grid barriers

  const dim3 lgrid(2 * NWGD), lblk(256);

  // ---- layer 0: input GEMMs (WMMA) then recurrence ----
  for (int d = 0; d < 2; ++d)
    run_gemm(stream, states, NH, Wih0 + (size_t)d * NG * NH, NH, 0,
             b0 + (size_t)d * NG, Xg + (size_t)d * 1024 * NG, NG, 1024, NG, NH);
  lstm_recur<<<lgrid, lblk, 0, stream>>>(Xg, Whh0, Y0, hgl, sync + 0, 0);

  // ---- layer 1 ----
  for (int d = 0; d < 2; ++d)
    run_gemm(stream, Y0, NH2, Wih12 + (size_t)(0 * 2 + d) * NG * NH2, NH2, 0,
             b12 + (size_t)(0 * 2 + d) * NG, Xg + (size_t)d * 1024 * NG, NG,
             1024, NG, NH2);
  lstm_recur<<<lgrid, lblk, 0, stream>>>(Xg, Whh12 + (size_t)0 * 2 * NG * NH,
                                         Y1, hgl, sync + 1, 0);

  // ---- layer 2 (tanh applied to output) ----
  for (int d = 0; d < 2; ++d)
    run_gemm(stream, Y1, NH2, Wih12 + (size_t)(1 * 2 + d) * NG * NH2, NH2, 0,
             b12 + (size_t)(1 * 2 + d) * NG, Xg + (size_t)d * 1024 * NG, NG,
             1024, NG, NH2);
  lstm_recur<<<lgrid, lblk, 0, stream>>>(Xg, Whh12 + (size_t)1 * 2 * NG * NH,
                                         Y0, hgl, sync + 2, 1);

  // ---- attention projections (WMMA): hq = actions@W1q.T, hk = out@W1k.T ----
  run_gemm(stream, actions, NH, attnW1, 3 * NH, 0,  nullptr, hq, NH, NB * NS2, NH, NH);
  run_gemm(stream, Y0,      NH2, attnW1, 3 * NH, NH, nullptr, hk, NH, NB * NS1, NH, NH2);

  attn_scores<<<NB * NS2, 256, 0, stream>>>(hq, hk, attnb1, attnW2, attnb2,
                                            amask, smask, P);
  states_feat_k<<<NB * NS2, 256, 0, stream>>>(P, Y0, SF);

  // ---- head: tmp = SF @ weight.T (WMMA), then contract with actions ----
  run_gemm(stream, SF, NH2, weight, NH2, 0, nullptr, tmp, NLAB * NH,
           NB * NS2, NLAB * NH, NH2);
  adv_reduce<<<NB * NS2, 256, 0, stream>>>(tmp, actions, bias, adv);
  finalize_k<<<NB, 256, 0, stream>>>(SF, adv, amask, valW, valb, out);
}